// LocalAttnDecoderRNN_47768626266274
// MI455X (gfx1250) — compile-verified
//
#include <hip/hip_runtime.h>
#include <math.h>

#define Bn   64
#define Sn   2048
#define Hn   512
#define Vn   50257
#define NBLK 32
#define Ln   64        // S / NBLK

typedef __attribute__((ext_vector_type(2))) float v2f;
typedef __attribute__((ext_vector_type(8))) float v8f;

// ---------------------------------------------------------------------------
// q[b,h] = sum_g hidden1[b,g] * attn_W[g,h] ;  cb[b] = hidden1[b,:] . attn_b
// ---------------------------------------------------------------------------
__global__ void k_query(const float* __restrict__ hid1,
                        const float* __restrict__ attn_W,
                        const float* __restrict__ attn_b,
                        float* __restrict__ q, float* __restrict__ cb) {
    __shared__ float h1s[Hn];
    __shared__ float red[256];
    const int b = blockIdx.x, t = threadIdx.x;
    h1s[t]       = hid1[b * Hn + t];
    h1s[t + 256] = hid1[b * Hn + t + 256];
    __syncthreads();
    // two output columns per thread
    for (int h = t; h < Hn; h += 256) {
        float acc = 0.f;
        for (int g = 0; g < Hn; ++g) acc += h1s[g] * attn_W[g * Hn + h];
        q[b * Hn + h] = acc;
    }
    float p = h1s[t] * attn_b[t] + h1s[t + 256] * attn_b[t + 256];
    red[t] = p;
    __syncthreads();
    for (int s = 128; s > 0; s >>= 1) {
        if (t < s) red[t] += red[t + s];
        __syncthreads();
    }
    if (t == 0) cb[b] = red[0];
}

// ---------------------------------------------------------------------------
// Per-(b,n) block attention: stage 64x512 tile in LDS, scores, softmax,
// weighted sum.  attn_w written to d_out, block context to ws.
// ---------------------------------------------------------------------------
#define TPAD 516   // 512 + 4 : bank-conflict pad, keeps float4 alignment
__global__ void k_block_attn(const float* __restrict__ eo,
                             const float* __restrict__ q,
                             const float* __restrict__ cb,
                             float* __restrict__ attn_w_out,
                             float* __restrict__ bctx) {
    extern __shared__ float tile[];          // 64 * 516 floats
    __shared__ float qs[Hn];
    __shared__ float sw[Ln];
    __shared__ float s_max, s_sum;
    const int n = blockIdx.x, b = blockIdx.y, t = threadIdx.x;

    qs[t]       = q[b * Hn + t];
    qs[t + 256] = q[b * Hn + t + 256];

    // cooperative tile load: 8192 float4, 32 per thread
    const float* src = eo + ((size_t)b * Sn + n * Ln) * Hn;
#pragma unroll
    for (int i = 0; i < 32; ++i) {
        int f4  = t + 256 * i;
        int row = f4 >> 7;          // 128 float4 per row
        int c4  = f4 & 127;
        float4 v = *(const float4*)(src + row * Hn + c4 * 4);
        *(float4*)(&tile[row * TPAD + c4 * 4]) = v;
    }
    __syncthreads();

    if (t < Ln) {                   // scores: one row per thread
        float acc = 0.f;
        const float* r = &tile[t * TPAD];
        for (int k = 0; k < Hn; ++k) acc += qs[k] * r[k];
        sw[t] = acc + cb[b];
    }
    __syncthreads();
    if (t == 0) {
        float m = sw[0];
        for (int l = 1; l < Ln; ++l) m = fmaxf(m, sw[l]);
        s_max = m;
    }
    __syncthreads();
    if (t < Ln) sw[t] = __expf(sw[t] - s_max);
    __syncthreads();
    if (t == 0) {
        float s = 0.f;
        for (int l = 0; l < Ln; ++l) s += sw[l];
        s_sum = s;
    }
    __syncthreads();
    if (t < Ln) {
        float w = sw[t] / s_sum;
        sw[t] = w;
        attn_w_out[((size_t)b * NBLK + n) * Ln + t] = w;
    }
    __syncthreads();
    for (int h = t; h < Hn; h += 256) {     // weighted sum over rows
        float acc = 0.f;
        for (int l = 0; l < Ln; ++l) acc += sw[l] * tile[l * TPAD + h];
        bctx[((size_t)b * NBLK + n) * Hn + h] = acc;
    }
}

// ---------------------------------------------------------------------------
// context[b,h] = sum_n baw[b,n]*bctx[b,n,h]; build xcat=[emb|ctx], xcat2 ctx
// ---------------------------------------------------------------------------
__global__ void k_context(const float* __restrict__ bctx,
                          const float* __restrict__ baw,
                          const float* __restrict__ emb,
                          const int* __restrict__ ids,
                          float* __restrict__ ctx_out,
                          float* __restrict__ xcat,
                          float* __restrict__ xcat2) {
    const int b = blockIdx.x, t = threadIdx.x;
    const int id = ids[b];
    for (int h = t; h < Hn; h += 256) {
        float c = 0.f;
        for (int n = 0; n < NBLK; ++n)
            c += baw[b * NBLK + n] * bctx[((size_t)b * NBLK + n) * Hn + h];
        ctx_out[b * Hn + h]       = c;
        xcat[b * 2 * Hn + Hn + h] = c;
        xcat2[b * 2 * Hn + Hn + h] = c;
        xcat[b * 2 * Hn + h] = emb[(size_t)id * Hn + h];
    }
}

// ---------------------------------------------------------------------------
// Generic NT GEMM on V_WMMA_F32_16X16X4_F32:  Out[m,n] = X[m,:K].W[n,:K]+bias
// M fixed = 64.  256 threads = 8 waves: wave -> (mtile = w&3, ntile = w>>2).
// K chunked by 64 through padded LDS (row stride 68).
// ---------------------------------------------------------------------------
#define LDA 68
__global__ void k_gemm_wmma(const float* __restrict__ X, int ldx,
                            const float* __restrict__ W,
                            const float* __restrict__ bias,
                            float* __restrict__ Out, int ldo,
                            int N, int K) {
    __shared__ float As[64 * LDA];
    __shared__ float Bs[32 * LDA];
    const int t    = threadIdx.x;
    const int lane = t & 31;
    const int wav  = t >> 5;
    const int mt   = (wav & 3) * 16;
    const int nt   = (wav >> 2) * 16;
    const int n0   = blockIdx.x * 32;

    v8f acc = {};
    const int nk = K >> 6;
    for (int kc = 0; kc < nk; ++kc) {
        const int k0 = kc << 6;
        __syncthreads();
        // A chunk: 64 rows x 16 float4 -> 4 f4/thread
#pragma unroll
        for (int i = 0; i < 4; ++i) {
            int f4  = t + 256 * i;
            int row = f4 >> 4, c4 = f4 & 15;
            float4 v = *(const float4*)(X + (size_t)row * ldx + k0 + c4 * 4);
            *(float4*)(&As[row * LDA + c4 * 4]) = v;
        }
        // B chunk: 32 rows x 16 float4 -> 2 f4/thread (zero OOB rows)
#pragma unroll
        for (int i = 0; i < 2; ++i) {
            int f4  = t + 256 * i;
            int row = f4 >> 4, c4 = f4 & 15;
            float4 v = make_float4(0.f, 0.f, 0.f, 0.f);
            if (n0 + row < N)
                v = *(const float4*)(W + (size_t)(n0 + row) * K + k0 + c4 * 4);
            *(float4*)(&Bs[row * LDA + c4 * 4]) = v;
        }
        __syncthreads();

        const int lm   = lane & 15;
        const int koff = (lane < 16) ? 0 : 2;    // ISA A/B 16x4 lane layout
#pragma unroll
        for (int k4 = 0; k4 < 16; ++k4) {
            const int kk = k4 * 4 + koff;
            v2f a, bfr;
            a[0]   = As[(mt + lm) * LDA + kk];
            a[1]   = As[(mt + lm) * LDA + kk + 1];
            bfr[0] = Bs[(nt + lm) * LDA + kk];
            bfr[1] = Bs[(nt + lm) * LDA + kk + 1];
            acc = __builtin_amdgcn_wmma_f32_16x16x4_f32(
                false, a, false, bfr, (short)0, acc, false, false);
        }
    }

    const int gcol = n0 + nt + (lane & 15);
    if (gcol < N) {
        const float bv = bias ? bias[gcol] : 0.f;
#pragma unroll
        for (int r = 0; r < 8; ++r) {
            const int grow = mt + ((lane < 16) ? r : r + 8);
            Out[(size_t)grow * ldo + gcol] = acc[r] + bv;
        }
    }
}

// ---------------------------------------------------------------------------
// GRU gate pointwise:  h' = (1-z)*n + z*h
// ---------------------------------------------------------------------------
__device__ __forceinline__ float sigm(float x) { return 1.f / (1.f + __expf(-x)); }

__global__ void k_gru_gate(const float* __restrict__ gx,
                           const float* __restrict__ gh,
                           const float* __restrict__ hprev,
                           float* __restrict__ hnew,
                           float* __restrict__ hnew2) {
    const int idx = blockIdx.x * 256 + threadIdx.x;   // B*H = 32768
    if (idx >= Bn * Hn) return;
    const int b = idx >> 9, i = idx & 511;
    const float xr = gx[b * 3 * Hn + i];
    const float xz = gx[b * 3 * Hn + Hn + i];
    const float xn = gx[b * 3 * Hn + 2 * Hn + i];
    const float hr = gh[b * 3 * Hn + i];
    const float hz = gh[b * 3 * Hn + Hn + i];
    const float hn = gh[b * 3 * Hn + 2 * Hn + i];
    const float r = sigm(xr + hr);
    const float z = sigm(xz + hz);
    const float n = tanhf(xn + r * hn);
    const float h = hprev[b * Hn + i];
    const float o = (1.f - z) * n + z * h;
    hnew[b * Hn + i] = o;
    if (hnew2) hnew2[b * 2 * Hn + i] = o;   // out_in = [h1 | context]
}

// ---------------------------------------------------------------------------
// per-row max & log-sum-exp over V, then apply log-softmax in place
// ---------------------------------------------------------------------------
__global__ void k_lse(const float* __restrict__ logits,
                      float* __restrict__ rmax, float* __restrict__ rlog) {
    __shared__ float red[256];
    const int b = blockIdx.x, t = threadIdx.x;
    const float* row = logits + (size_t)b * Vn;
    float m = -3.4e38f;
    for (int v = t; v < Vn; v += 256) m = fmaxf(m, row[v]);
    red[t] = m; __syncthreads();
    for (int s = 128; s > 0; s >>= 1) {
        if (t < s) red[t] = fmaxf(red[t], red[t + s]);
        __syncthreads();
    }
    const float mx = red[0]; __syncthreads();
    float sum = 0.f;
    for (int v = t; v < Vn; v += 256) sum += __expf(row[v] - mx);
    red[t] = sum; __syncthreads();
    for (int s = 128; s > 0; s >>= 1) {
        if (t < s) red[t] += red[t + s];
        __syncthreads();
    }
    if (t == 0) { rmax[b] = mx; rlog[b] = logf(red[0]); }
}

__global__ void k_lsm_apply(float* __restrict__ logits,
                            const float* __restrict__ rmax,
                            const float* __restrict__ rlog) {
    const size_t idx = (size_t)blockIdx.x * 256 + threadIdx.x;
    if (idx >= (size_t)Bn * Vn) return;
    const int b = (int)(idx / Vn);
    logits[idx] = logits[idx] - rmax[b] - rlog[b];
}

// ---------------------------------------------------------------------------
extern "C" void kernel_launch(void* const* d_in, const int* in_sizes, int n_in,
                              void* d_out, int out_size, void* d_ws, size_t ws_size,
                              hipStream_t stream) {
    const int*   ids    = (const int*)  d_in[0];
    const float* hidden = (const float*)d_in[1];           // [2,B,H]
    const float* baw    = (const float*)d_in[2];           // [B,32]
    const float* eo     = (const float*)d_in[3];           // [B,S,H]
    const float* emb    = (const float*)d_in[5];
    const float* attn_W = (const float*)d_in[6];
    const float* attn_b = (const float*)d_in[7];
    const float* W_ih0  = (const float*)d_in[8];
    const float* W_hh0  = (const float*)d_in[9];
    const float* b_ih0  = (const float*)d_in[10];
    const float* b_hh0  = (const float*)d_in[11];
    const float* W_ih1  = (const float*)d_in[12];
    const float* W_hh1  = (const float*)d_in[13];
    const float* b_ih1  = (const float*)d_in[14];
    const float* b_hh1  = (const float*)d_in[15];
    const float* out_W  = (const float*)d_in[16];
    const float* out_b  = (const float*)d_in[17];

    const float* hid0 = hidden;
    const float* hid1 = hidden + Bn * Hn;

    float* out = (float*)d_out;
    float* log_probs = out;                                 // [B,V]
    float* newhid0   = out + (size_t)Bn * Vn;               // [B,H]
    float* newhid1   = newhid0 + Bn * Hn;                   // [B,H]
    float* ctx_out   = newhid1 + Bn * Hn;                   // [B,H]
    float* attnw_out = ctx_out + Bn * Hn;                   // [B,32,64]

    float* ws    = (float*)d_ws;
    float* q     = ws;                 // B*H
    float* cb    = q     + Bn * Hn;    // B
    float* bctx  = cb    + Bn;         // B*32*H
    float* xcat  = bctx  + (size_t)Bn * NBLK * Hn;  // B*2H
    float* xcat2 = xcat  + Bn * 2 * Hn;             // B*2H
    float* gx0   = xcat2 + Bn * 2 * Hn;             // B*3H
    float* gh0   = gx0   + Bn * 3 * Hn;
    float* gx1   = gh0   + Bn * 3 * Hn;
    float* gh1   = gx1   + Bn * 3 * Hn;
    float* rmax  = gh1   + Bn * 3 * Hn;             // B
    float* rlog  = rmax  + Bn;                      // B

    // 1. fused attention query
    k_query<<<Bn, 256, 0, stream>>>(hid1, attn_W, attn_b, q, cb);

    // 2. block attention (132 KB dynamic LDS tile)
    k_block_attn<<<dim3(NBLK, Bn), 256, Ln * TPAD * sizeof(float), stream>>>(
        eo, q, cb, attnw_out, bctx);

    // 3. context + embedding concat
    k_context<<<Bn, 256, 0, stream>>>(bctx, baw, emb, ids, ctx_out, xcat, xcat2);

    // 4. GRU layer 0
    k_gemm_wmma<<<(3 * Hn) / 32, 256, 0, stream>>>(xcat, 2 * Hn, W_ih0, b_ih0,
                                                   gx0, 3 * Hn, 3 * Hn, 2 * Hn);
    k_gemm_wmma<<<(3 * Hn) / 32, 256, 0, stream>>>(hid0, Hn, W_hh0, b_hh0,
                                                   gh0, 3 * Hn, 3 * Hn, Hn);
    k_gru_gate<<<(Bn * Hn) / 256, 256, 0, stream>>>(gx0, gh0, hid0, newhid0, nullptr);

    // 5. GRU layer 1  (h1 also written into out_in = [h1 | context])
    k_gemm_wmma<<<(3 * Hn) / 32, 256, 0, stream>>>(newhid0, Hn, W_ih1, b_ih1,
                                                   gx1, 3 * Hn, 3 * Hn, Hn);
    k_gemm_wmma<<<(3 * Hn) / 32, 256, 0, stream>>>(hid1, Hn, W_hh1, b_hh1,
                                                   gh1, 3 * Hn, 3 * Hn, Hn);
    k_gru_gate<<<(Bn * Hn) / 256, 256, 0, stream>>>(gx1, gh1, hid1, newhid1, xcat2);

    // 6. output projection (memory-bound fp32 WMMA GEMM, 206 MB stream)
    k_gemm_wmma<<<(Vn + 31) / 32, 256, 0, stream>>>(xcat2, 2 * Hn, out_W, out_b,
                                                    log_probs, Vn, Vn, 2 * Hn);

    // 7. log-softmax over V
    k_lse<<<Bn, 256, 0, stream>>>(log_probs, rmax, rlog);
    k_lsm_apply<<<(int)(((size_t)Bn * Vn + 255) / 256), 256, 0, stream>>>(
        log_probs, rmax, rlog);
}